// RNAFeatures_24051816857780
// MI455X (gfx1250) — compile-verified
//
#include <hip/hip_runtime.h>
#include <math.h>

// Problem constants (match reference)
#define Bc 4
#define Nc 2048
#define Kc 30
#define BNc (Bc * Nc)
#define BNKc (Bc * Nc * Kc)
#define NFc 128
#define NODE_IN 117
#define EDGE_IN 115
#define NODE_KP 120   // 117 padded to multiple of 4
#define EDGE_KP 116   // 115 padded to multiple of 4
#define FMAXV 3.402823466e+38f

typedef __attribute__((ext_vector_type(2))) float v2f;
typedef __attribute__((ext_vector_type(8))) float v8f;

__device__ __forceinline__ float sgnf(float x) {
    return (x > 0.f) ? 1.f : ((x < 0.f) ? -1.f : 0.f);
}
__device__ __forceinline__ void cross3(const float* a, const float* b, float* o) {
    o[0] = a[1]*b[2] - a[2]*b[1];
    o[1] = a[2]*b[0] - a[0]*b[2];
    o[2] = a[0]*b[1] - a[1]*b[0];
}
__device__ __forceinline__ float dot3(const float* a, const float* b) {
    return a[0]*b[0] + a[1]*b[1] + a[2]*b[2];
}
// _safe_normalize: where(n>0, v/n, 0)
__device__ __forceinline__ void safen3(const float* v, float* o) {
    float n = sqrtf(dot3(v, v));
    if (n > 0.f) { o[0]=v[0]/n; o[1]=v[1]/n; o[2]=v[2]/n; }
    else         { o[0]=0.f;   o[1]=0.f;   o[2]=0.f; }
}
// _f_normalize: v / max(n, 1e-12)
__device__ __forceinline__ void fnorm3(const float* v, float* o) {
    float n = fmaxf(sqrtf(dot3(v, v)), 1e-12f);
    o[0]=v[0]/n; o[1]=v[1]/n; o[2]=v[2]/n;
}

// ---------------------------------------------------------------------------
// Kernel 1: kNN over P atoms. One block per (b,i). D row in LDS, row max,
// then 30 iterative argmins (lowest-index tie-break, like lax.top_k).
// ---------------------------------------------------------------------------
__global__ __launch_bounds__(256) void knn_kernel(const float* __restrict__ X,
                                                  const float* __restrict__ mask,
                                                  int* __restrict__ Eidx) {
    __shared__ float sD[Nc];
    __shared__ float sval[256];
    __shared__ int   sidx[256];
    const int row = blockIdx.x;              // b*N + i
    const int b = row / Nc, i = row - b * Nc;
    const float* Xb = X + (size_t)b * Nc * 18;
    const float pix = Xb[i*18+0], piy = Xb[i*18+1], piz = Xb[i*18+2];
    const float mi = mask[row];
    const int t = threadIdx.x;

    float lmax = -FMAXV;
    for (int j = t; j < Nc; j += 256) {
        float dx = Xb[j*18+0]-pix, dy = Xb[j*18+1]-piy, dz = Xb[j*18+2]-piz;
        float m2 = mi * mask[b*Nc + j];
        float d  = (1.f - m2) * 10000.f + m2 * sqrtf(dx*dx + dy*dy + dz*dz + 1e-6f);
        sD[j] = d;
        lmax = fmaxf(lmax, d);
    }
    sval[t] = lmax;
    __syncthreads();
    for (int s = 128; s > 0; s >>= 1) {
        if (t < s) sval[t] = fmaxf(sval[t], sval[t+s]);
        __syncthreads();
    }
    const float Dmax = sval[0];
    __syncthreads();
    for (int j = t; j < Nc; j += 256) {
        float m2 = mi * mask[b*Nc + j];
        sD[j] += (1.f - m2) * (Dmax + 1.f);
    }
    __syncthreads();

    for (int k = 0; k < Kc; ++k) {
        float bv = FMAXV; int bi = 0x7FFFFFFF;
        #pragma unroll
        for (int q = 0; q < Nc/256; ++q) {       // blocked scan: ascending j
            int j = t * (Nc/256) + q;
            float v = sD[j];
            if (v < bv) { bv = v; bi = j; }
        }
        sval[t] = bv; sidx[t] = bi;
        __syncthreads();
        for (int s = 128; s > 0; s >>= 1) {
            if (t < s) {
                float v2 = sval[t+s]; int i2 = sidx[t+s];
                if (v2 < sval[t] || (v2 == sval[t] && i2 < sidx[t])) {
                    sval[t] = v2; sidx[t] = i2;
                }
            }
            __syncthreads();
        }
        int win = sidx[0];
        if (t == 0) { Eidx[row*Kc + k] = win; sD[win] = FMAXV; }
        __syncthreads();
    }
}

// ---------------------------------------------------------------------------
// Kernel 2: contact count on C4 + fused 1->32->16 MLP
// ---------------------------------------------------------------------------
__global__ __launch_bounds__(256) void contact_kernel(const float* __restrict__ X,
                                                      const float* __restrict__ cW1,
                                                      const float* __restrict__ cb1,
                                                      const float* __restrict__ cW2,
                                                      const float* __restrict__ cb2,
                                                      float* __restrict__ cfeat) {
    __shared__ float scnt[256];
    __shared__ float sh[32];
    const int row = blockIdx.x;
    const int b = row / Nc, i = row - b * Nc;
    const float* Xb = X + (size_t)b * Nc * 18;
    const float cx = Xb[i*18+9], cy = Xb[i*18+10], cz = Xb[i*18+11];  // atom 3 = C4
    const int t = threadIdx.x;

    float cnt = 0.f;
    for (int j = t; j < Nc; j += 256) {
        float dx = Xb[j*18+9]-cx, dy = Xb[j*18+10]-cy, dz = Xb[j*18+11]-cz;
        float d2 = dx*dx + dy*dy + dz*dz;
        if (j != i && d2 <= 64.f) cnt += 1.f;
    }
    scnt[t] = cnt;
    __syncthreads();
    for (int s = 128; s > 0; s >>= 1) {
        if (t < s) scnt[t] += scnt[t+s];
        __syncthreads();
    }
    float c = scnt[0];
    if (t < 32) sh[t] = fmaxf(0.f, c * cW1[t] + cb1[t]);
    __syncthreads();
    if (t < 16) {
        float o = cb2[t];
        #pragma unroll
        for (int u = 0; u < 32; ++u) o += sh[u] * cW2[u*16 + t];
        cfeat[(size_t)row*16 + t] = o;
    }
}

// ---------------------------------------------------------------------------
// Kernel 3: per-node local frame Q (rows: b1, n0, b1 x n0); last node zero
// ---------------------------------------------------------------------------
__global__ __launch_bounds__(256) void qframe_kernel(const float* __restrict__ X,
                                                     float* __restrict__ Qw) {
    const int g = blockIdx.x * 256 + threadIdx.x;
    if (g >= BNc) return;
    const int b = g / Nc, t = g - b * Nc;
    float q[9] = {0,0,0,0,0,0,0,0,0};
    if (t < Nc - 1) {
        const float* Xb = X + (size_t)b * Nc * 18;
        float d0[3], d1[3], u0[3], u1[3], tmp[3], n0[3], b1[3], r2[3];
        #pragma unroll
        for (int c = 0; c < 3; ++c) {
            float c3 = Xb[t*18 + 12 + c];
            float o3 = Xb[t*18 + 15 + c];
            float pn = Xb[(t+1)*18 + c];
            d0[c] = o3 - c3;
            d1[c] = pn - o3;
        }
        safen3(d0, u0);
        safen3(d1, u1);
        cross3(u0, u1, tmp); safen3(tmp, n0);
        tmp[0]=u0[0]-u1[0]; tmp[1]=u0[1]-u1[1]; tmp[2]=u0[2]-u1[2];
        safen3(tmp, b1);
        cross3(b1, n0, r2);
        q[0]=b1[0]; q[1]=b1[1]; q[2]=b1[2];
        q[3]=n0[0]; q[4]=n0[1]; q[5]=n0[2];
        q[6]=r2[0]; q[7]=r2[1]; q[8]=r2[2];
    }
    #pragma unroll
    for (int c = 0; c < 9; ++c) Qw[(size_t)g*9 + c] = q[c];
}

// ---------------------------------------------------------------------------
// Kernel 4: node features -> hV (padded to 120)
// ---------------------------------------------------------------------------
__global__ __launch_bounds__(256) void nodefeat_kernel(const float* __restrict__ X,
                                                       const float* __restrict__ Qw,
                                                       const float* __restrict__ cfeat,
                                                       float* __restrict__ hV) {
    const int g = blockIdx.x * 256 + threadIdx.x;
    if (g >= BNc) return;
    const int b = g / Nc, i = g - b * Nc;
    const float* Xb = X + (size_t)b * Nc * 18;   // X6 flat: Xb[s*3+c]
    float* out = hV + (size_t)g * NODE_KP;

    // --- dihedrals (12) : lag-5 chain ---
    for (int c = 0; c < 6; ++c) {
        int m = 6*i + c;
        float Dv = 0.f;
        if (m >= 3 && m <= 6*Nc - 5) {
            int tt = m - 3;
            float U[3][3], tmp[3], n2[3], n1[3];
            #pragma unroll
            for (int s = 0; s < 3; ++s) {
                int si = tt + s;
                float d[3];
                #pragma unroll
                for (int cc = 0; cc < 3; ++cc)
                    d[cc] = Xb[(si+5)*3 + cc] - Xb[si*3 + cc];
                fnorm3(d, U[s]);
            }
            cross3(U[0], U[1], tmp); fnorm3(tmp, n2);
            cross3(U[1], U[2], tmp); fnorm3(tmp, n1);
            float cosD = dot3(n2, n1);
            cosD = fminf(fmaxf(cosD, -1.f + 1e-7f), 1.f - 1e-7f);
            Dv = sgnf(dot3(U[0], n1)) * acosf(cosD);
        }
        out[c]     = cosf(Dv);
        out[6 + c] = sinf(Dv);
    }

    // --- node RBFs (80): atoms O5,C5,C4,C3,O3 vs P ---
    const float px = Xb[i*18+0], py = Xb[i*18+1], pz = Xb[i*18+2];
    for (int a = 1; a < 6; ++a) {
        float dx = Xb[i*18 + a*3 + 0] - px;
        float dy = Xb[i*18 + a*3 + 1] - py;
        float dz = Xb[i*18 + a*3 + 2] - pz;
        float d = sqrtf(dx*dx + dy*dy + dz*dz + 1e-6f);
        #pragma unroll
        for (int r = 0; r < 16; ++r) {
            float mu = (20.f / 15.f) * (float)r;
            float z = (d - mu) * 0.8f;               // sigma = 1.25
            out[12 + 16*(a-1) + r] = expf(-z*z);
        }
    }

    // --- V_direct (9): rotate (P,C5,C4) - C3 into local frame, safe-normalize
    float q[9];
    #pragma unroll
    for (int c = 0; c < 9; ++c) q[c] = Qw[(size_t)g*9 + c];
    float xc[3] = { Xb[i*18+12], Xb[i*18+13], Xb[i*18+14] };
    const int ats[3] = {0, 2, 3};
    for (int p = 0; p < 3; ++p) {
        float v[3], w[3], o[3];
        #pragma unroll
        for (int c = 0; c < 3; ++c) v[c] = Xb[i*18 + ats[p]*3 + c] - xc[c];
        #pragma unroll
        for (int r = 0; r < 3; ++r) w[r] = q[3*r]*v[0] + q[3*r+1]*v[1] + q[3*r+2]*v[2];
        safen3(w, o);
        out[92 + 3*p + 0] = o[0];
        out[92 + 3*p + 1] = o[1];
        out[92 + 3*p + 2] = o[2];
    }

    // --- contact MLP features (16) + pad ---
    #pragma unroll
    for (int v = 0; v < 16; ++v) out[101 + v] = cfeat[(size_t)g*16 + v];
    out[117] = 0.f; out[118] = 0.f; out[119] = 0.f;
}

// ---------------------------------------------------------------------------
// Shared GEMM + LayerNorm body: 16 rows x 128 cols, 8 waves, f32 WMMA 16x16x4
// ---------------------------------------------------------------------------
template <int KP, int KTRUE>
__device__ __forceinline__ void gemm_ln_tile(const float sA[16][KP],
                                             float sO[16][NFc],
                                             float sMu[16], float sSg[16],
                                             const float* __restrict__ W,
                                             const float* __restrict__ bias,
                                             const float* __restrict__ gain,
                                             const float* __restrict__ bvec,
                                             float* __restrict__ out,
                                             int rowBase, int t) {
    const int lane = t & 31;
    const int w    = t >> 5;                 // wave id 0..7 -> 16-col tile
    const int m    = lane & 15;              // A row / D col within tile
    const int kh   = (lane >> 4) << 1;       // K half offset {0,2}
    const int col  = (w << 4) + m;           // global output column
    v8f acc = {0.f,0.f,0.f,0.f,0.f,0.f,0.f,0.f};
    #pragma unroll 4
    for (int kk = 0; kk < KP/4; ++kk) {
        const int kb = kk * 4;
        const int k0 = kb + kh, k1 = k0 + 1;
        v2f a, bb;
        a.x  = sA[m][k0];
        a.y  = sA[m][k1];
        bb.x = (k0 < KTRUE) ? W[k0*NFc + col] : 0.f;
        bb.y = (k1 < KTRUE) ? W[k1*NFc + col] : 0.f;
        acc = __builtin_amdgcn_wmma_f32_16x16x4_f32(
            false, a, false, bb, (short)0, acc, false, false);
    }
    const int half = lane >> 4;
    #pragma unroll
    for (int r = 0; r < 8; ++r)
        sO[r + half*8][col] = acc[r] + bias[col];
    __syncthreads();

    if (t < 16) {                            // two-pass stats, ddof=1
        float s1 = 0.f;
        for (int c = 0; c < NFc; ++c) s1 += sO[t][c];
        float mu = s1 / (float)NFc;
        float s2 = 0.f;
        for (int c = 0; c < NFc; ++c) { float d = sO[t][c] - mu; s2 += d*d; }
        sMu[t] = mu;
        sSg[t] = sqrtf(s2 / (float)(NFc - 1) + 1e-6f);
    }
    __syncthreads();
    for (int idx = t; idx < 16*NFc; idx += 256) {
        int r = idx >> 7, c = idx & 127;
        out[(size_t)(rowBase + r)*NFc + c] =
            gain[c] * (sO[r][c] - sMu[r]) / (sSg[r] + 1e-6f) + bvec[c];
    }
}

// ---------------------------------------------------------------------------
// Kernel 5: node GEMM (hV @ node_W + b) + LayerNorm, WMMA f32 16x16x4
// ---------------------------------------------------------------------------
__global__ __launch_bounds__(256) void node_gemm_kernel(const float* __restrict__ hV,
                                                        const float* __restrict__ W,
                                                        const float* __restrict__ bias,
                                                        const float* __restrict__ gain,
                                                        const float* __restrict__ bvec,
                                                        float* __restrict__ out) {
    __shared__ float sA[16][NODE_KP];
    __shared__ float sO[16][NFc];
    __shared__ float sMu[16], sSg[16];
    const int rowBase = blockIdx.x * 16;
    const int t = threadIdx.x;
    for (int idx = t; idx < 16*NODE_KP; idx += 256) {
        int r = idx / NODE_KP, c = idx - r*NODE_KP;
        sA[r][c] = hV[(size_t)(rowBase + r)*NODE_KP + c];
    }
    __syncthreads();
    gemm_ln_tile<NODE_KP, NODE_IN>(sA, sO, sMu, sSg, W, bias, gain, bvec, out, rowBase, t);
}

// ---------------------------------------------------------------------------
// Kernel 6: fused edge features + GEMM + LayerNorm (16 edges per block)
// ---------------------------------------------------------------------------
__global__ __launch_bounds__(256) void edge_kernel(const float* __restrict__ X,
                                                   const int* __restrict__ Eidx,
                                                   const float* __restrict__ Qw,
                                                   const float* __restrict__ W,
                                                   const float* __restrict__ bias,
                                                   const float* __restrict__ gain,
                                                   const float* __restrict__ bvec,
                                                   float* __restrict__ out) {
    __shared__ float sA[16][EDGE_KP];
    __shared__ float sO[16][NFc];
    __shared__ float sMu[16], sSg[16];
    const int e0 = blockIdx.x * 16;
    const int t = threadIdx.x;

    if (t < 16) {
        const int e   = e0 + t;
        const int b   = e / (Nc * Kc);
        const int rem = e - b * (Nc * Kc);
        const int i   = rem / Kc;
        const int j   = Eidx[e];
        const float* Xb = X + (size_t)b * Nc * 18;
        const float* Xi = Xb + i * 18;
        const float* Xj = Xb + j * 18;
        const float* qi = Qw + (size_t)(b*Nc + i) * 9;
        const float* qj = Qw + (size_t)(b*Nc + j) * 9;
        float* f = &sA[t][0];

        // E_orient (4): quaternion of R = Qi^T * Qj
        float R[3][3];
        #pragma unroll
        for (int ii = 0; ii < 3; ++ii)
            #pragma unroll
            for (int ll = 0; ll < 3; ++ll)
                R[ii][ll] = qi[ii]*qj[ll] + qi[3+ii]*qj[3+ll] + qi[6+ii]*qj[6+ll];
        float Rxx = R[0][0], Ryy = R[1][1], Rzz = R[2][2];
        float mx = 0.5f * sqrtf(fabsf(1.f + Rxx - Ryy - Rzz));
        float my = 0.5f * sqrtf(fabsf(1.f - Rxx + Ryy - Rzz));
        float mz = 0.5f * sqrtf(fabsf(1.f - Rxx - Ryy + Rzz));
        float qx = sgnf(R[2][1] - R[1][2]) * mx;
        float qy = sgnf(R[0][2] - R[2][0]) * my;
        float qz = sgnf(R[1][0] - R[0][1]) * mz;
        float qw = 0.5f * sqrtf(fmaxf(0.f, 1.f + Rxx + Ryy + Rzz));
        float nn = fmaxf(sqrtf(qx*qx + qy*qy + qz*qz + qw*qw), 1e-12f);
        f[0] = qx/nn; f[1] = qy/nn; f[2] = qz/nn; f[3] = qw/nn;

        // E_dist (96): atoms 0..5 of i vs P of j
        for (int a = 0; a < 6; ++a) {
            float dx = Xi[a*3+0] - Xj[0];
            float dy = Xi[a*3+1] - Xj[1];
            float dz = Xi[a*3+2] - Xj[2];
            float d = sqrtf(dx*dx + dy*dy + dz*dz + 1e-6f);
            #pragma unroll
            for (int r = 0; r < 16; ++r) {
                float mu = (20.f / 15.f) * (float)r;
                float z = (d - mu) * 0.8f;
                f[4 + a*16 + r] = expf(-z*z);
            }
        }

        // E_direct (15): neighbor atoms {P,O5,C5,C4,O3} - C3_i, rotated by Qi
        const int ats[5] = {0, 1, 2, 3, 5};
        for (int p = 0; p < 5; ++p) {
            float v[3], wv[3], o[3];
            #pragma unroll
            for (int c = 0; c < 3; ++c) v[c] = Xj[ats[p]*3 + c] - Xi[12 + c];
            #pragma unroll
            for (int r = 0; r < 3; ++r)
                wv[r] = qi[3*r]*v[0] + qi[3*r+1]*v[1] + qi[3*r+2]*v[2];
            safen3(wv, o);
            f[100 + 3*p + 0] = o[0];
            f[100 + 3*p + 1] = o[1];
            f[100 + 3*p + 2] = o[2];
        }
        f[115] = 0.f;   // pad
    }
    __syncthreads();
    gemm_ln_tile<EDGE_KP, EDGE_IN>(sA, sO, sMu, sSg, W, bias, gain, bvec, out, e0, t);
}

// ---------------------------------------------------------------------------
// Kernel 7: E_sparse (2, B*N*K) int32
// ---------------------------------------------------------------------------
__global__ __launch_bounds__(256) void esparse_kernel(const int* __restrict__ Eidx,
                                                      int* __restrict__ out) {
    const int e = blockIdx.x * 256 + threadIdx.x;
    if (e >= BNKc) return;
    const int b = e / (Nc * Kc);
    const int i = (e - b * (Nc * Kc)) / Kc;
    out[e]        = b * Nc + i;          // dst
    out[BNKc + e] = b * Nc + Eidx[e];    // src
}

// ---------------------------------------------------------------------------
extern "C" void kernel_launch(void* const* d_in, const int* in_sizes, int n_in,
                              void* d_out, int out_size, void* d_ws, size_t ws_size,
                              hipStream_t stream) {
    const float* X      = (const float*)d_in[0];
    /* d_in[1] = S (unused by reference outputs) */
    const float* mask   = (const float*)d_in[2];
    const float* node_W = (const float*)d_in[3];
    const float* node_b = (const float*)d_in[4];
    const float* edge_W = (const float*)d_in[5];
    const float* edge_b = (const float*)d_in[6];
    const float* gain_n = (const float*)d_in[7];
    const float* bias_n = (const float*)d_in[8];
    const float* gain_e = (const float*)d_in[9];
    const float* bias_e = (const float*)d_in[10];
    const float* cW1    = (const float*)d_in[11];
    const float* cb1    = (const float*)d_in[12];
    const float* cW2    = (const float*)d_in[13];
    const float* cb2    = (const float*)d_in[14];
    (void)in_sizes; (void)n_in; (void)out_size; (void)ws_size;

    // workspace layout
    char*  ws    = (char*)d_ws;
    int*   Eidx  = (int*)ws;                                   // BNK ints
    float* Qw    = (float*)(ws + (size_t)BNKc * sizeof(int));  // BN*9
    float* cfeat = Qw + (size_t)BNc * 9;                       // BN*16
    float* hV    = cfeat + (size_t)BNc * 16;                   // BN*120

    // output layout: _V | _E | E_sparse
    float* outV = (float*)d_out;
    float* outE = outV + (size_t)BNc * NFc;
    int*   outS = (int*)(outE + (size_t)BNKc * NFc);

    knn_kernel    <<<BNc, 256, 0, stream>>>(X, mask, Eidx);
    contact_kernel<<<BNc, 256, 0, stream>>>(X, cW1, cb1, cW2, cb2, cfeat);
    qframe_kernel <<<(BNc + 255)/256, 256, 0, stream>>>(X, Qw);
    nodefeat_kernel<<<(BNc + 255)/256, 256, 0, stream>>>(X, Qw, cfeat, hV);
    node_gemm_kernel<<<BNc/16, 256, 0, stream>>>(hV, node_W, node_b, gain_n, bias_n, outV);
    edge_kernel   <<<BNKc/16, 256, 0, stream>>>(X, Eidx, Qw, edge_W, edge_b,
                                                gain_e, bias_e, outE);
    esparse_kernel<<<(BNKc + 255)/256, 256, 0, stream>>>(Eidx, outS);
}